// AttentionInput_35450660061740
// MI455X (gfx1250) — compile-verified
//
#include <hip/hip_runtime.h>
#include <math.h>

#define B_ 32
#define S_ 4096
#define D_ 256

typedef __attribute__((ext_vector_type(2))) float v2f;
typedef __attribute__((ext_vector_type(4))) float v4f;
typedef __attribute__((ext_vector_type(8))) float v8f;

// ---------------------------------------------------------------------------
// Kernel 1: entity scores A1[b,s] = wd[b,s,:]·e1[b,:], A2 likewise, via
// V_WMMA_F32_16X16X4_F32. One workgroup = 1 batch x 128 consecutive s rows,
// 8 waves, each wave owns 16 rows. wd tiles are DMA'd into a double-buffered
// LDS tile with GLOBAL_LOAD_ASYNC_TO_LDS_B128 (ASYNCcnt), overlapping the
// next chunk's copy with the current chunk's WMMAs.
// B fragment: even columns carry e1, odd columns carry e2 (redundant columns
// are free — the WMMA computes 16 columns anyway). Column 0 = A1, col 1 = A2.
// grid = (S/128, B), block = 256.
// ---------------------------------------------------------------------------
__global__ __launch_bounds__(256) void scores_wmma_kernel(
    const float* __restrict__ wd, const float* __restrict__ e1,
    const float* __restrict__ e2, float* __restrict__ A1,
    float* __restrict__ A2) {
  __shared__ float tile[2][128][68];  // double buffer; stride 68: 16B-aligned
                                      // rows (272B) and 68%64==4 kills the
                                      // 16-row bank-conflict pattern
  __shared__ float se1[D_];
  __shared__ float se2[D_];

  const int b  = blockIdx.y;
  const int s0 = blockIdx.x * 128;
  const int t  = threadIdx.x;

  // Stage the two entity vectors for this batch.
  se1[t] = e1[b * D_ + t];
  se2[t] = e2[b * D_ + t];

  const int wave = t >> 5;
  const int lane = t & 31;
  const int h    = lane >> 4;          // half-wave select (ISA VGPR layouts)
  const int n    = lane & 15;          // C column / A row-within-half
  const int rowA = wave * 16 + n;      // tile row this lane supplies to A
  const float* sel = (n & 1) ? se2 : se1;  // one pointer select, no masks

  // Issue one 64-float-wide chunk of the 128-row tile as 8 async 16B copies
  // per thread: 16 lanes cover one 256B row segment (coalesced), 256 threads
  // cover 16 rows per pass.
  auto issue_chunk = [&](int chunkIdx, int buf) {
    const int d0 = chunkIdx * 64;
#pragma unroll
    for (int p = 0; p < 8; ++p) {
      const int row = p * 16 + (t >> 4);
      const int c4  = (t & 15) * 4;
      const unsigned lds = (unsigned)(size_t)&tile[buf][row][c4];
      const unsigned long long ga = (unsigned long long)(
          wd + (size_t)(b * S_ + s0 + row) * D_ + d0 + c4);
      asm volatile("global_load_async_to_lds_b128 %0, %1, off th:TH_LOAD_NT"
                   :: "v"(lds), "v"(ga)
                   : "memory");
    }
  };

  v8f c = {0.f, 0.f, 0.f, 0.f, 0.f, 0.f, 0.f, 0.f};

  issue_chunk(0, 0);  // prologue: chunk 0 -> buffer 0

  for (int chunk = 0; chunk < 4; ++chunk) {
    const int buf = chunk & 1;
    // My DMAs for this chunk are done...
    asm volatile("s_wait_asynccnt 0x0" ::: "memory");
    // ...and after the barrier, everyone's are (and everyone has finished
    // computing on the buffer we are about to refill).
    __syncthreads();
    if (chunk < 3) issue_chunk(chunk + 1, buf ^ 1);

    const int d0 = chunk * 64;
#pragma unroll
    for (int k = 0; k < 64; k += 4) {
      // A fragment (16x4 f32): VGPR v, half h -> K = v + 2h, M = lane%16.
      v2f a;
      a.x = tile[buf][rowA][k + 2 * h];
      a.y = tile[buf][rowA][k + 2 * h + 1];
      // B fragment (4x16 f32): VGPR v, half h -> K = v + 2h, N = lane%16.
      const int dk = d0 + k + 2 * h;
      v2f bm;
      bm.x = sel[dk];
      bm.y = sel[dk + 1];
      // D = A*B + C  (8 args: neg_a, A, neg_b, B, c_mod, C, reuse_a, reuse_b)
      c = __builtin_amdgcn_wmma_f32_16x16x4_f32(false, a, false, bm,
                                                (short)0, c, false, false);
    }
  }

  // C layout: lane holds M = v + 8h at column N = lane%16. Column 0 is A1
  // (e1 in even B columns), column 1 is A2.
  if (n < 2) {
    float* dst = (n == 0) ? A1 : A2;
    const int base = b * S_ + s0 + wave * 16 + 8 * h;
#pragma unroll
    for (int v = 0; v < 8; ++v) dst[base + v] = c[v];
  }
}

// ---------------------------------------------------------------------------
// Kernel 2: per-batch softmax over S=4096 for both score sets, fused into
// alpha = 0.5*(softmax(A1)^2 + softmax(A2)^2).  grid = B, block = 256.
// A1/A2 are 512 KB each -> L2 resident; rereads are free.
// ---------------------------------------------------------------------------
__global__ __launch_bounds__(256) void softmax_alpha_kernel(
    const float* __restrict__ A1, const float* __restrict__ A2,
    float* __restrict__ alpha) {
  __shared__ float r1[256];
  __shared__ float r2[256];
  const int b = blockIdx.x;
  const int t = threadIdx.x;
  const float* a1 = A1 + (size_t)b * S_;
  const float* a2 = A2 + (size_t)b * S_;

  float m1 = -INFINITY, m2 = -INFINITY;
  for (int s = t; s < S_; s += 256) {
    m1 = fmaxf(m1, a1[s]);
    m2 = fmaxf(m2, a2[s]);
  }
  r1[t] = m1; r2[t] = m2;
  __syncthreads();
  for (int off = 128; off > 0; off >>= 1) {
    if (t < off) {
      r1[t] = fmaxf(r1[t], r1[t + off]);
      r2[t] = fmaxf(r2[t], r2[t + off]);
    }
    __syncthreads();
  }
  m1 = r1[0]; m2 = r2[0];
  __syncthreads();

  float s1 = 0.f, s2 = 0.f;
  for (int s = t; s < S_; s += 256) {
    s1 += __expf(a1[s] - m1);
    s2 += __expf(a2[s] - m2);
  }
  r1[t] = s1; r2[t] = s2;
  __syncthreads();
  for (int off = 128; off > 0; off >>= 1) {
    if (t < off) {
      r1[t] += r1[t + off];
      r2[t] += r2[t + off];
    }
    __syncthreads();
  }
  const float inv1 = 1.f / r1[0];
  const float inv2 = 1.f / r2[0];

  for (int s = t; s < S_; s += 256) {
    const float x1 = __expf(a1[s] - m1) * inv1;
    const float x2 = __expf(a2[s] - m2) * inv2;
    alpha[(size_t)b * S_ + s] = 0.5f * (x1 * x1 + x2 * x2);
  }
}

// ---------------------------------------------------------------------------
// Kernel 3: out = wM * alpha[..,None]. Pure stream (128 MiB in, 128 MiB out):
// 16B non-temporal load/store so it doesn't pollute L2; alpha (512 KB)
// stays regular-temporal and hits L2.
// ---------------------------------------------------------------------------
__global__ __launch_bounds__(256) void scale_kernel(
    const float* __restrict__ wM, const float* __restrict__ alpha,
    float* __restrict__ out) {
  const size_t i4 = (size_t)blockIdx.x * 256 + threadIdx.x;  // vec4 index
  const size_t bs = (i4 * 4) / D_;  // D=256 is a multiple of 4: one s per vec
  const float a = alpha[bs];
  v4f v = __builtin_nontemporal_load(reinterpret_cast<const v4f*>(wM) + i4);
  v *= a;
  __builtin_nontemporal_store(v, reinterpret_cast<v4f*>(out) + i4);
}

// ---------------------------------------------------------------------------
extern "C" void kernel_launch(void* const* d_in, const int* in_sizes, int n_in,
                              void* d_out, int out_size, void* d_ws,
                              size_t ws_size, hipStream_t stream) {
  const float* wM = (const float*)d_in[0];
  const float* wd = (const float*)d_in[1];
  const float* e1 = (const float*)d_in[2];
  const float* e2 = (const float*)d_in[3];
  float* out = (float*)d_out;

  float* ws    = (float*)d_ws;
  float* A1    = ws;                          // B*S floats
  float* A2    = ws + (size_t)B_ * S_;        // B*S floats
  float* alpha = ws + 2 * (size_t)B_ * S_;    // B*S floats  (1.5 MB total)

  dim3 g1(S_ / 128, B_);
  scores_wmma_kernel<<<g1, 256, 0, stream>>>(wd, e1, e2, A1, A2);
  softmax_alpha_kernel<<<B_, 256, 0, stream>>>(A1, A2, alpha);

  const size_t n4 = (size_t)B_ * S_ * D_ / 4;  // 8,388,608 vec4s
  scale_kernel<<<(unsigned)(n4 / 256), 256, 0, stream>>>(wM, alpha, out);
}